// PointNeXtSegModel_33079838113821
// MI455X (gfx1250) — compile-verified
//
#include <hip/hip_runtime.h>
#include <hip/hip_bf16.h>

typedef __attribute__((ext_vector_type(2))) float v2f;
typedef __attribute__((ext_vector_type(8))) float v8f;

#define NBATCH 4

// ---------------------------------------------------------------------------
// Generic fp32 WMMA GEMM:  C[M,Nout] = act(A[M,K4] * Wp + bias)
// Wp is pair-packed: Wp[((k>>1)*Np + n)*2 + (k&1)] = W[k][n]  (zero-padded)
// K4 multiple of 4, Np multiple of 16, M multiple of 16.
// Block = 128 threads (4 waves); each wave -> 16 rows x NT*16 cols.
// NT is a compile-time constant so accumulators stay in fixed VGPRs.
// ---------------------------------------------------------------------------
template <int NT>
__global__ void gemm_wmma_f32(const float* __restrict__ A,
                              const float* __restrict__ Wp,
                              float* __restrict__ C,
                              const float* __restrict__ bias,
                              int M, int K4, int Np, int Nout, int relu)
{
    const int wave = threadIdx.x >> 5;
    const int lane = threadIdx.x & 31;
    const int row0 = (blockIdx.x * 4 + wave) * 16;
    const int col0 = blockIdx.y * (NT * 16);
    if (row0 >= M) return;

    const int m    = lane & 15;          // row within A-tile / col within B-tile
    const int koff = (lane >> 4) << 1;   // K sub-offset per ISA A/B layout

    v8f acc[NT] = {};

    const float* arow = A + (size_t)(row0 + m) * K4 + koff;
    for (int k = 0; k < K4; k += 4) {
        v2f a = *(const v2f*)(arow + k);
        // pair index for (k+koff, k+koff+1); k%4==0, koff even
        const float* wp = Wp + (size_t)((k + koff) >> 1) * Np * 2;
#pragma unroll
        for (int t = 0; t < NT; ++t) {
            int n = col0 + t * 16 + m;
            v2f b = *(const v2f*)(wp + (size_t)n * 2);
            acc[t] = __builtin_amdgcn_wmma_f32_16x16x4_f32(
                false, a, false, b, (short)0, acc[t], false, false);
        }
    }

    const int rbase = row0 + (lane >> 4) * 8;   // C/D layout: VGPR r -> M=r (+8 hi half)
#pragma unroll
    for (int t = 0; t < NT; ++t) {
        int n = col0 + t * 16 + m;
        if (n >= Nout) continue;
        float bv = bias ? bias[n] : 0.0f;
#pragma unroll
        for (int r = 0; r < 8; ++r) {
            float v = acc[t][r] + bv;
            if (relu) v = fmaxf(v, 0.0f);
            C[(size_t)(rbase + r) * Nout + n] = v;
        }
    }
}

// ---------------------------------------------------------------------------
// Pack weight W[K,D] (row-major) into zero-padded, pair-packed Wp:
//   Wp[((k>>1)*Np + n)*2 + (k&1)] = W[k][n]
// ---------------------------------------------------------------------------
__global__ void pack_w_kernel(const float* __restrict__ W, float* __restrict__ Wp,
                              int K, int D, int K4, int Np)
{
    int g = blockIdx.x * blockDim.x + threadIdx.x;
    int total = K4 * Np;
    if (g >= total) return;
    int j = g & 1;
    int pn = g >> 1;
    int n = pn % Np;
    int p = pn / Np;
    int k = 2 * p + j;
    Wp[g] = (k < K && n < D) ? W[(size_t)k * D + n] : 0.0f;
}

// Pad rows: out[row*K4+c] = c<K ? in[row*K+c] : 0   (stem's 3->4 pad)
__global__ void pad_rows_kernel(const float* __restrict__ in, float* __restrict__ out,
                                int M, int K, int K4)
{
    int g = blockIdx.x * blockDim.x + threadIdx.x;
    if (g >= M * K4) return;
    int row = g / K4, c = g % K4;
    out[g] = (c < K) ? in[(size_t)row * K + c] : 0.0f;
}

// ---------------------------------------------------------------------------
// Farthest point sampling: one block per batch. dists kept in global scratch.
// Tie-break = lowest index (matches jnp.argmax first-max semantics).
// ---------------------------------------------------------------------------
__global__ void fps_kernel(const float* __restrict__ xyz, int N, int S,
                           int* __restrict__ fidx, float* __restrict__ dists)
{
    const int b = blockIdx.x;
    const float* p = xyz + (size_t)b * N * 3;
    int* idxs = fidx + b * S;
    float* dist = dists + (size_t)b * N;
    const int tid = threadIdx.x;
    const int nth = blockDim.x;

    __shared__ float s_val[256];
    __shared__ int   s_idx[256];
    __shared__ int   s_sel;

    for (int i = tid; i < N; i += nth) dist[i] = 1e10f;
    if (tid == 0) { idxs[0] = 0; s_sel = 0; }
    __syncthreads();

    int sel = 0;
    for (int s = 1; s < S; ++s) {
        float lx = p[sel * 3 + 0], ly = p[sel * 3 + 1], lz = p[sel * 3 + 2];
        float best = -1.0f; int bestI = 0x7fffffff;
        for (int i = tid; i < N; i += nth) {
            float dx = p[i * 3 + 0] - lx;
            float dy = p[i * 3 + 1] - ly;
            float dz = p[i * 3 + 2] - lz;
            float d = dx * dx + dy * dy + dz * dz;
            float dm = fminf(dist[i], d);
            dist[i] = dm;
            if (dm > best) { best = dm; bestI = i; }
        }
        s_val[tid] = best; s_idx[tid] = bestI;
        __syncthreads();
        for (int off = 128; off > 0; off >>= 1) {
            if (tid < off) {
                float ov = s_val[tid + off]; int oi = s_idx[tid + off];
                if (ov > s_val[tid] || (ov == s_val[tid] && oi < s_idx[tid])) {
                    s_val[tid] = ov; s_idx[tid] = oi;
                }
            }
            __syncthreads();
        }
        if (tid == 0) { s_sel = s_idx[0]; idxs[s] = s_sel; }
        __syncthreads();
        sel = s_sel;
    }
}

// ---------------------------------------------------------------------------
// Ball query: thread per center; first K indices (ascending) within radius,
// padded with the first found (reference pads with idx[:, :1]).
// ---------------------------------------------------------------------------
__global__ void ballq_kernel(const float* __restrict__ xyz,
                             const float* __restrict__ centers,
                             int N, int S, float r2, int K,
                             int* __restrict__ nidx, int total)
{
    int g = blockIdx.x * blockDim.x + threadIdx.x;
    if (g >= total) return;                 // total = B*S
    int b = g / S;
    const float* p = xyz + (size_t)b * N * 3;
    const float* c = centers + (size_t)g * 3;
    int* out = nidx + (size_t)g * K;
    float cx = c[0], cy = c[1], cz = c[2];
    int cnt = 0, first = 0;
    for (int i = 0; i < N && cnt < K; ++i) {
        float dx = p[3 * i + 0] - cx;
        float dy = p[3 * i + 1] - cy;
        float dz = p[3 * i + 2] - cz;
        float d2 = dx * dx + dy * dy + dz * dz;
        if (d2 <= r2) {
            if (cnt == 0) first = i;
            out[cnt++] = i;
        }
    }
    for (; cnt < K; ++cnt) out[cnt] = first;
}

// Gather rows: dst[bs, :] = src[b, idx[bs], :]   (used for new_xyz and shortcut feats)
__global__ void gather_rows_kernel(const float* __restrict__ src,
                                   const int* __restrict__ idx,
                                   float* __restrict__ dst,
                                   int N, int S, int C, int total)
{
    int g = blockIdx.x * blockDim.x + threadIdx.x;
    if (g >= total) return;                 // total = B*S*C
    int c = g % C, bs = g / C, b = bs / S;
    int i = idx[bs];
    dst[g] = src[((size_t)b * N + i) * C + c];
}

// ---------------------------------------------------------------------------
// Group + concat: A[(b,s,k), 0:3] = (xyz[nidx]-new_xyz)/r ; [3:3+C]=feats ; pad 0
// ---------------------------------------------------------------------------
__global__ void group_kernel(const float* __restrict__ xyz,
                             const float* __restrict__ feats,
                             const float* __restrict__ new_xyz,
                             const int* __restrict__ nidx,
                             int N, int S, int K, int C, int K4, float inv_r,
                             float* __restrict__ A, int total)
{
    int g = blockIdx.x * blockDim.x + threadIdx.x;
    if (g >= total) return;                 // total = B*S*K
    int b = g / (S * K);
    int sk = g % (S * K);
    int s = sk / K;
    int idx = nidx[g];
    const float* pt  = xyz + ((size_t)b * N + idx) * 3;
    const float* ctr = new_xyz + ((size_t)b * S + s) * 3;
    float* row = A + (size_t)g * K4;
    row[0] = (pt[0] - ctr[0]) * inv_r;
    row[1] = (pt[1] - ctr[1]) * inv_r;
    row[2] = (pt[2] - ctr[2]) * inv_r;
    const float* f = feats + ((size_t)b * N + idx) * C;
    for (int c = 0; c < C; ++c) row[3 + c] = f[c];
    for (int c = 3 + C; c < K4; ++c) row[c] = 0.0f;
}

// out[bs,d] = relu(max_k H[(bs*K+k), d] + sh[bs,d])
__global__ void maxpool_add_relu_kernel(const float* __restrict__ H,
                                        const float* __restrict__ sh,
                                        float* __restrict__ out,
                                        int K, int D, int total)
{
    int g = blockIdx.x * blockDim.x + threadIdx.x;
    if (g >= total) return;                 // total = B*S*D
    int bs = g / D, d = g % D;
    const float* hp = H + (size_t)bs * K * D + d;
    float m = hp[0];
    for (int k = 1; k < K; ++k) m = fmaxf(m, hp[(size_t)k * D]);
    float v = m + sh[g];
    out[g] = fmaxf(v, 0.0f);
}

// ---------------------------------------------------------------------------
// FP: 3-NN inverse-distance interpolation + skip concat -> A[(b,n), Cc+Cf]
// ---------------------------------------------------------------------------
__global__ void fp_interp_concat_kernel(const float* __restrict__ fine_xyz,
                                        const float* __restrict__ coarse_xyz,
                                        const float* __restrict__ skip_f,
                                        const float* __restrict__ coarse_f,
                                        int Nf, int Nc, int Cf, int Cc,
                                        float* __restrict__ A, int total)
{
    int g = blockIdx.x * blockDim.x + threadIdx.x;
    if (g >= total) return;                 // total = B*Nf
    int b = g / Nf;
    const float* fx = fine_xyz + (size_t)g * 3;
    const float* cx = coarse_xyz + (size_t)b * Nc * 3;
    float px = fx[0], py = fx[1], pz = fx[2];

    float d0 = 1e30f, d1 = 1e30f, d2v = 1e30f;
    int i0 = 0, i1 = 0, i2 = 0;
    for (int i = 0; i < Nc; ++i) {
        float dx = px - cx[3 * i + 0];
        float dy = py - cx[3 * i + 1];
        float dz = pz - cx[3 * i + 2];
        float d = dx * dx + dy * dy + dz * dz;
        if (d < d0)       { d2v = d1; i2 = i1; d1 = d0; i1 = i0; d0 = d; i0 = i; }
        else if (d < d1)  { d2v = d1; i2 = i1; d1 = d;  i1 = i; }
        else if (d < d2v) { d2v = d;  i2 = i; }
    }
    float w0 = 1.0f / (d0 + 1e-8f);
    float w1 = 1.0f / (d1 + 1e-8f);
    float w2 = 1.0f / (d2v + 1e-8f);
    float ws = w0 + w1 + w2;
    w0 /= ws; w1 /= ws; w2 /= ws;

    float* row = A + (size_t)g * (Cc + Cf);
    const float* cf = coarse_f + (size_t)b * Nc * Cc;
    const float* r0 = cf + (size_t)i0 * Cc;
    const float* r1 = cf + (size_t)i1 * Cc;
    const float* r2 = cf + (size_t)i2 * Cc;
    for (int c = 0; c < Cc; ++c)
        row[c] = w0 * r0[c] + w1 * r1[c] + w2 * r2[c];
    const float* sf = skip_f + (size_t)g * Cf;
    for (int c = 0; c < Cf; ++c) row[Cc + c] = sf[c];
}

// ---------------------------------------------------------------------------
// Host orchestration
// ---------------------------------------------------------------------------
extern "C" void kernel_launch(void* const* d_in, const int* in_sizes, int n_in,
                              void* d_out, int out_size, void* d_ws, size_t ws_size,
                              hipStream_t stream)
{
    (void)in_sizes; (void)n_in; (void)out_size; (void)ws_size;
    const int B = NBATCH;
    const int Nlev[5] = {16384, 1024, 256, 64, 16};
    const float RADII[4] = {0.1f, 0.2f, 0.4f, 0.8f};
    const int KSAMP = 32;
    const int ENC[5] = {32, 64, 128, 256, 512};

    // ---- inputs (pytree flatten order) ----
    const float* in_xyz   = (const float*)d_in[0];
    const float* in_feat  = (const float*)d_in[1];
    const float* stem_w   = (const float*)d_in[2];
    const float* saW1[4], *saW2[4], *saWsh[4];
    for (int i = 0; i < 4; ++i) {
        saW1[i]  = (const float*)d_in[3 + 3 * i];
        saW2[i]  = (const float*)d_in[4 + 3 * i];
        saWsh[i] = (const float*)d_in[5 + 3 * i];
    }
    const float* fpW1[4], *fpW2[4];
    for (int j = 0; j < 4; ++j) {
        fpW1[j] = (const float*)d_in[15 + 2 * j];
        fpW2[j] = (const float*)d_in[16 + 2 * j];
    }
    const float* head_w1 = (const float*)d_in[23];
    const float* head_b1 = (const float*)d_in[24];
    const float* head_w2 = (const float*)d_in[25];
    const float* head_b2 = (const float*)d_in[26];
    float* out = (float*)d_out;

    // ---- scratch arena ----
    char* wsb = (char*)d_ws;
    size_t off = 0;
    auto allocf = [&](size_t n) -> float* {
        float* p = (float*)(wsb + off);
        off = (off + n * sizeof(float) + 255) & ~(size_t)255;
        return p;
    };
    auto alloci = [&](size_t n) -> int* {
        int* p = (int*)(wsb + off);
        off = (off + n * sizeof(int) + 255) & ~(size_t)255;
        return p;
    };

    float* flev[5];
    flev[0] = allocf((size_t)B * 16384 * 32);
    flev[1] = allocf((size_t)B * 1024 * 64);
    flev[2] = allocf((size_t)B * 256 * 128);
    flev[3] = allocf((size_t)B * 64 * 256);
    flev[4] = allocf((size_t)B * 16 * 512);
    const float* xyzp[5];
    xyzp[0] = in_xyz;
    float* xyzbuf[4];
    for (int i = 0; i < 4; ++i) { xyzbuf[i] = allocf((size_t)B * Nlev[i + 1] * 3); xyzp[i + 1] = xyzbuf[i]; }

    int*   fidx  = alloci((size_t)B * 1024);
    int*   nidx  = alloci((size_t)B * 1024 * KSAMP);
    float* dists = allocf((size_t)B * 16384);
    float* sh_in  = allocf((size_t)B * 1024 * 32);    // max B*S*C = 131072
    float* sh_out = allocf((size_t)B * 1024 * 64);    // max 262144

    float* BUF_A  = allocf(6291456);                  // max: FP0 concat 65536*96
    float* BUF_H1 = allocf(8388608);                  // max: SA0 hidden 131072*64
    float* BUF_H2 = allocf(8388608);

    float* decf[4];                                   // decoder outputs, index = fine level
    decf[3] = allocf((size_t)B * 64 * 256);
    decf[2] = allocf((size_t)B * 256 * 128);
    decf[1] = allocf((size_t)B * 1024 * 64);
    decf[0] = allocf((size_t)B * 16384 * 32);

    // packed weights
    auto rnd4 = [](int k) { return (k + 3) & ~3; };
    float* stemWp = allocf(4 * 32);
    float* saW1p[4], *saW2p[4], *saWshp[4];
    int saK4[4];
    for (int i = 0; i < 4; ++i) {
        saK4[i] = rnd4(3 + ENC[i]);
        saW1p[i]  = allocf((size_t)saK4[i] * ENC[i + 1]);
        saW2p[i]  = allocf((size_t)ENC[i + 1] * ENC[i + 1]);
        saWshp[i] = allocf((size_t)ENC[i] * ENC[i + 1]);
    }
    float* fpW1p[4], *fpW2p[4];
    for (int j = 0; j < 4; ++j) {
        int f = 3 - j;
        int kin = ENC[f + 1] + ENC[f];
        fpW1p[j] = allocf((size_t)kin * ENC[f]);
        fpW2p[j] = allocf((size_t)ENC[f] * ENC[f]);
    }
    float* headW1p = allocf(32 * 64);
    float* headW2p = allocf(64 * 16);

    // ---- launch helpers ----
    auto pack = [&](const float* W, float* Wp, int K, int D, int K4, int Np) {
        int total = K4 * Np;
        pack_w_kernel<<<(total + 255) / 256, 256, 0, stream>>>(W, Wp, K, D, K4, Np);
    };
    auto gemm = [&](const float* Ap, const float* Wp, float* Cp, const float* bias,
                    int M, int K4, int Np, int Nout, int relu) {
        // NT uniform across grid: Np is 16, 32, or a multiple of 64
        if (Np >= 64) {
            dim3 g((M + 63) / 64, Np / 64);
            gemm_wmma_f32<4><<<g, 128, 0, stream>>>(Ap, Wp, Cp, bias, M, K4, Np, Nout, relu);
        } else if (Np == 32) {
            dim3 g((M + 63) / 64, 1);
            gemm_wmma_f32<2><<<g, 128, 0, stream>>>(Ap, Wp, Cp, bias, M, K4, Np, Nout, relu);
        } else {
            dim3 g((M + 63) / 64, 1);
            gemm_wmma_f32<1><<<g, 128, 0, stream>>>(Ap, Wp, Cp, bias, M, K4, Np, Nout, relu);
        }
    };

    // ---- pack all weights ----
    pack(stem_w, stemWp, 3, 32, 4, 32);
    for (int i = 0; i < 4; ++i) {
        pack(saW1[i],  saW1p[i],  3 + ENC[i], ENC[i + 1], saK4[i],     ENC[i + 1]);
        pack(saW2[i],  saW2p[i],  ENC[i + 1], ENC[i + 1], ENC[i + 1],  ENC[i + 1]);
        pack(saWsh[i], saWshp[i], ENC[i],     ENC[i + 1], ENC[i],      ENC[i + 1]);
    }
    for (int j = 0; j < 4; ++j) {
        int f = 3 - j;
        int kin = ENC[f + 1] + ENC[f];
        pack(fpW1[j], fpW1p[j], kin, ENC[f], kin, ENC[f]);
        pack(fpW2[j], fpW2p[j], ENC[f], ENC[f], ENC[f], ENC[f]);
    }
    pack(head_w1, headW1p, 32, 64, 32, 64);
    pack(head_w2, headW2p, 64, 13, 64, 16);

    // ---- stem: f0 = relu(features @ stem_w) ----
    {
        int M = B * Nlev[0];
        int tot = M * 4;
        pad_rows_kernel<<<(tot + 255) / 256, 256, 0, stream>>>(in_feat, BUF_A, M, 3, 4);
        gemm(BUF_A, stemWp, flev[0], nullptr, M, 4, 32, 32, 1);
    }

    // ---- encoder: 4 residual SA blocks ----
    for (int i = 0; i < 4; ++i) {
        int N = Nlev[i], S = Nlev[i + 1], C = ENC[i], D = ENC[i + 1];
        float r = RADII[i];
        int K4 = saK4[i];

        fps_kernel<<<B, 256, 0, stream>>>(xyzp[i], N, S, fidx, dists);

        int totC3 = B * S * 3;
        gather_rows_kernel<<<(totC3 + 255) / 256, 256, 0, stream>>>(
            xyzp[i], fidx, xyzbuf[i], N, S, 3, totC3);

        int totBS = B * S;
        ballq_kernel<<<(totBS + 255) / 256, 256, 0, stream>>>(
            xyzp[i], xyzbuf[i], N, S, r * r, KSAMP, nidx, totBS);

        int totSh = B * S * C;
        gather_rows_kernel<<<(totSh + 255) / 256, 256, 0, stream>>>(
            flev[i], fidx, sh_in, N, S, C, totSh);

        int totG = B * S * KSAMP;
        group_kernel<<<(totG + 255) / 256, 256, 0, stream>>>(
            xyzp[i], flev[i], xyzbuf[i], nidx, N, S, KSAMP, C, K4, 1.0f / r, BUF_A, totG);

        int Mrows = B * S * KSAMP;
        gemm(BUF_A,  saW1p[i], BUF_H1, nullptr, Mrows, K4, D, D, 1);   // relu(x @ W1)
        gemm(BUF_H1, saW2p[i], BUF_H2, nullptr, Mrows, D,  D, D, 0);   // x @ W2
        gemm(sh_in, saWshp[i], sh_out, nullptr, B * S, C, D, D, 0);    // shortcut

        int totO = B * S * D;
        maxpool_add_relu_kernel<<<(totO + 255) / 256, 256, 0, stream>>>(
            BUF_H2, sh_out, flev[i + 1], KSAMP, D, totO);
    }

    // ---- decoder: 4 FP blocks ----
    const float* fcur = flev[4];
    for (int j = 0; j < 4; ++j) {
        int f = 3 - j;
        int Nf = Nlev[f], Nc = Nlev[f + 1];
        int Cf = ENC[f], Cc = ENC[f + 1];
        int kin = Cc + Cf;

        int totNf = B * Nf;
        fp_interp_concat_kernel<<<(totNf + 255) / 256, 256, 0, stream>>>(
            xyzp[f], xyzp[f + 1], flev[f], fcur, Nf, Nc, Cf, Cc, BUF_A, totNf);

        gemm(BUF_A,  fpW1p[j], BUF_H1,  nullptr, totNf, kin, Cf, Cf, 1);
        gemm(BUF_H1, fpW2p[j], decf[f], nullptr, totNf, Cf,  Cf, Cf, 1);
        fcur = decf[f];
    }

    // ---- head ----
    {
        int M = B * Nlev[0];
        gemm(decf[0], headW1p, BUF_H1, head_b1, M, 32, 64, 64, 1);
        gemm(BUF_H1, headW2p, out, head_b2, M, 64, 16, 13, 0);
    }
}